// ArrangementsContinuousDecoderRNN_8108898255129
// MI455X (gfx1250) — compile-verified
//
#include <hip/hip_runtime.h>
#include <math.h>

// ---------------------------------------------------------------------------
// ArrangementsContinuousDecoderRNN for MI455X (gfx1250, wave32, WMMA).
// fp32 end-to-end with V_WMMA_F32_16X16X4_F32 (latency-bound workload; exact
// numerics, matrix pipe engaged). LDS-tiled GEMM with async-to-LDS staging
// (GLOBAL_LOAD_ASYNC_TO_LDS_B128, ASYNCcnt); fused GRU step kernel that
// applies the gates directly on the WMMA accumulators.
// ---------------------------------------------------------------------------

namespace {
constexpr int kB = 64;     // batch
constexpr int kT = 128;    // text length
constexpr int kL = 24;     // clip length
constexpr int kH = 512;    // hidden
constexpr int kKC = 32;    // K-slice per LDS stage
constexpr int kLdsStride = 36;  // floats; 144B rows (16B aligned, bank-friendly)
}

typedef float v2f __attribute__((ext_vector_type(2)));
typedef float v4f __attribute__((ext_vector_type(4)));
typedef float v8f __attribute__((ext_vector_type(8)));
typedef int   v4i __attribute__((ext_vector_type(4)));

__device__ __forceinline__ float sigmoidf_(float x) { return 1.0f / (1.0f + expf(-x)); }

// ---------------------------------------------------------------------------
// Async global -> LDS staging (CDNA5 GLOBAL_LOAD_ASYNC_TO_LDS_B128, ASYNCcnt),
// with a synchronous fallback if the builtin is not available.
// Builtin signature (from hipcc diagnostic): (v4i as1* src, v4i as3* dst,
// imm int offset, imm int cpol).
// ---------------------------------------------------------------------------
#if __has_builtin(__builtin_amdgcn_global_load_async_to_lds_b128)
#define GRU_HAVE_ASYNC_LDS 1
#else
#define GRU_HAVE_ASYNC_LDS 0
#endif

#if GRU_HAVE_ASYNC_LDS
typedef __attribute__((address_space(1))) v4i* gv4i_p;
typedef __attribute__((address_space(3))) v4i* lv4i_p;
#endif

__device__ __forceinline__ void cp16_to_lds(const float* g, float* l) {
#if GRU_HAVE_ASYNC_LDS
  // generic->AS1: generic global address == global address.
  // generic->AS3: low 32 bits of the generic LDS address are the LDS offset.
  __builtin_amdgcn_global_load_async_to_lds_b128(
      (gv4i_p)(uintptr_t)g, (lv4i_p)(uint32_t)(uintptr_t)l, 0, 0);
#else
  *(v4f*)l = *(const v4f*)g;
#endif
}

__device__ __forceinline__ void stage_wait() {
#if GRU_HAVE_ASYNC_LDS
#if __has_builtin(__builtin_amdgcn_s_wait_asynccnt)
  __builtin_amdgcn_s_wait_asynccnt(0);
#else
  asm volatile("s_wait_asynccnt 0" ::: "memory");
#endif
#endif
}

// Copy a (rows x 32) fp32 tile from row-major gbase (leading dim K) starting
// at column kBase into LDS with row stride kLdsStride. rows*8 % 256 == 0 at
// every call site, so the loop is wave-uniform (EXEC stays all-ones).
__device__ __forceinline__ void stage_tile(const float* __restrict__ gbase,
                                           int rows, int K, int kBase,
                                           float* __restrict__ lds, int tid) {
  const int nf4 = rows * 8;  // 32 floats = 8 float4 per row
  for (int i = tid; i < nf4; i += 256) {
    const int r = i >> 3, c = (i & 7) * 4;
    cp16_to_lds(gbase + (size_t)r * K + kBase + c, lds + r * kLdsStride + c);
  }
}

// ---------------------------------------------------------------------------
// C(M,N) = A(M,K) @ W(N,K)^T + bias(N), fp32 WMMA 16x16x4, LDS-tiled.
// Block 256 threads = 8 waves in a 4(M) x 2(N) grid; each wave computes a
// 16x64 strip (four 16x16 accumulators). Macro-tile 64x128 per block.
// Requires M%64==0, N%128==0, K%32==0 (true for all call sites).
// ---------------------------------------------------------------------------
__global__ void gemm_bias_kernel(const float* __restrict__ A,
                                 const float* __restrict__ W,
                                 const float* __restrict__ bias,
                                 float* __restrict__ C,
                                 int M, int N, int K) {
  __shared__ float shA[64 * kLdsStride];
  __shared__ float shW[128 * kLdsStride];
  const int tid = threadIdx.x;
  const int lane = tid & 31, wave = tid >> 5;
  const int wm = wave & 3, wn = wave >> 2;
  const int mBase = blockIdx.y * 64;
  const int nBase = blockIdx.x * 128;
  const int half = lane >> 4, l16 = lane & 15;
  (void)M;

  v8f acc[4] = {};
  for (int k0 = 0; k0 < K; k0 += kKC) {
    stage_tile(A + (size_t)mBase * K, 64, K, k0, shA, tid);
    stage_tile(W + (size_t)nBase * K, 128, K, k0, shW, tid);
    stage_wait();
    __syncthreads();

    const float* aRow = shA + (wm * 16 + l16) * kLdsStride + 2 * half;
#pragma unroll
    for (int k = 0; k < kKC; k += 4) {
      v2f a = *(const v2f*)(aRow + k);
#pragma unroll
      for (int nt = 0; nt < 4; ++nt) {
        const float* wRow = shW + (wn * 64 + nt * 16 + l16) * kLdsStride + 2 * half;
        v2f b = *(const v2f*)(wRow + k);
        acc[nt] = __builtin_amdgcn_wmma_f32_16x16x4_f32(
            false, a, false, b, (short)0, acc[nt], false, false);
      }
    }
    __syncthreads();
  }

#pragma unroll
  for (int nt = 0; nt < 4; ++nt) {
    const int n = nBase + wn * 64 + nt * 16 + l16;
    const float bv = bias ? bias[n] : 0.0f;
#pragma unroll
    for (int v = 0; v < 8; ++v) {
      C[(size_t)(mBase + wm * 16 + v + 8 * half) * N + n] = acc[nt][v] + bv;
    }
  }
}

// ---------------------------------------------------------------------------
// Fused GRU step: gh = h_in @ Whh^T (+bhh) computed with WMMA, then the gate
// nonlinearity is applied directly on the accumulator registers and h_new is
// written out. Block bc owns hidden columns [bc*128, bc*128+128); it computes
// the r, z, n tiles for those columns (rows g*512+col of Whh), so the whole
// update is local to the block. grid = H/128 = 4 blocks, 256 threads.
// h is ping-ponged between h_in and h_out to avoid cross-block races.
// ---------------------------------------------------------------------------
__global__ void gru_step_kernel(const float* __restrict__ xp,
                                const float* __restrict__ h_in,
                                float* __restrict__ h_out,
                                const float* __restrict__ Whh,
                                const float* __restrict__ bhh,
                                float* __restrict__ ys,
                                int S, int t, int ys_stride, int col_off) {
  __shared__ float shA[64 * kLdsStride];
  __shared__ float shW[3][128 * kLdsStride];
  const int tid = threadIdx.x;
  const int lane = tid & 31, wave = tid >> 5;
  const int wm = wave & 3, wn = wave >> 2;
  const int nBase = blockIdx.x * 128;   // hidden-column slice
  const int half = lane >> 4, l16 = lane & 15;

  v8f acc[3][4] = {};
  for (int k0 = 0; k0 < kH; k0 += kKC) {
    stage_tile(h_in, 64, kH, k0, shA, tid);
#pragma unroll
    for (int g = 0; g < 3; ++g) {
      stage_tile(Whh + (size_t)(g * kH + nBase) * kH, 128, kH, k0, shW[g], tid);
    }
    stage_wait();
    __syncthreads();

    const float* aRow = shA + (wm * 16 + l16) * kLdsStride + 2 * half;
#pragma unroll
    for (int k = 0; k < kKC; k += 4) {
      v2f a = *(const v2f*)(aRow + k);
#pragma unroll
      for (int g = 0; g < 3; ++g) {
#pragma unroll
        for (int nt = 0; nt < 4; ++nt) {
          const float* wRow = shW[g] + (wn * 64 + nt * 16 + l16) * kLdsStride + 2 * half;
          v2f b = *(const v2f*)(wRow + k);
          acc[g][nt] = __builtin_amdgcn_wmma_f32_16x16x4_f32(
              false, a, false, b, (short)0, acc[g][nt], false, false);
        }
      }
    }
    __syncthreads();
  }

  // Gate math on the accumulator layout: acc[g][nt][v] holds the value for
  // batch row m = wm*16 + v + 8*half, hidden col n = nBase + wn*64 + nt*16 + l16.
#pragma unroll
  for (int nt = 0; nt < 4; ++nt) {
    const int n = nBase + wn * 64 + nt * 16 + l16;
    const float br = bhh[n], bz = bhh[kH + n], bn = bhh[2 * kH + n];
#pragma unroll
    for (int v = 0; v < 8; ++v) {
      const int m = wm * 16 + v + 8 * half;  // batch index
      const size_t xb = ((size_t)m * S + t) * (3 * kH);
      float r = sigmoidf_(xp[xb + n] + acc[0][nt][v] + br);
      float z = sigmoidf_(xp[xb + kH + n] + acc[1][nt][v] + bz);
      float nn = tanhf(xp[xb + 2 * kH + n] + r * (acc[2][nt][v] + bn));
      float hprev = h_in[(size_t)m * kH + n];
      float hnew = (1.0f - z) * nn + z * hprev;
      h_out[(size_t)m * kH + n] = hnew;
      ys[((size_t)m * S + t) * ys_stride + col_off + n] = hnew;
    }
  }
}

// ---------------------------------------------------------------------------
// Utility kernels
// ---------------------------------------------------------------------------
__global__ void fill_kernel(float* __restrict__ p, int n, float v) {
  int i = blockIdx.x * 256 + threadIdx.x;
  if (i < n) p[i] = v;
}

__global__ void gather_rows_kernel(const float* __restrict__ emb,
                                   const int* __restrict__ idx,
                                   float* __restrict__ out, int rows, int width) {
  int i = blockIdx.x * 256 + threadIdx.x;
  if (i >= rows * width) return;
  int r = i / width, c = i % width;
  out[i] = emb[(size_t)idx[r] * width + c];
}

__global__ void concat_kernel(const float* __restrict__ a, const float* __restrict__ b,
                              float* __restrict__ out, int rows, int wa, int wb) {
  int i = blockIdx.x * 256 + threadIdx.x;
  int w = wa + wb;
  if (i >= rows * w) return;
  int r = i / w, c = i % w;
  out[i] = (c < wa) ? a[(size_t)r * wa + c] : b[(size_t)r * wb + (c - wa)];
}

// ---------------------------------------------------------------------------
// Fused additive attention: one block per (b,l); never materializes the
// (B,L,T,H) tanh tensor (would be 400 MB).
// ---------------------------------------------------------------------------
__global__ void attn_kernel(const float* __restrict__ o_h,
                            const float* __restrict__ o_v,
                            const float* __restrict__ Wu,
                            const float* __restrict__ text_enc,
                            float* __restrict__ zout) {
  __shared__ float sh_oh[kH];
  __shared__ float sh_wu[kH];
  __shared__ float sh_sc[kT];
  const int tid = threadIdx.x;
  const int bl = blockIdx.x;
  const int b = bl / kL;
  for (int i = tid; i < kH; i += 256) {
    sh_oh[i] = o_h[(size_t)bl * kH + i];
    sh_wu[i] = Wu[i];
  }
  __syncthreads();

  const int wave = tid >> 5, lane = tid & 31;
  for (int t = wave; t < kT; t += 8) {
    const float* ov = o_v + ((size_t)b * kT + t) * kH;
    float acc = 0.0f;
    for (int hh = lane; hh < kH; hh += 32) acc += sh_wu[hh] * tanhf(sh_oh[hh] + ov[hh]);
    for (int off = 16; off > 0; off >>= 1) acc += __shfl_down(acc, off, 32);
    if (lane == 0) sh_sc[t] = acc;
  }
  __syncthreads();

  if (wave == 0) {  // softmax over T = 128 within wave 0 (4 scores per lane)
    float v[4];
    float m = -3.0e38f;
#pragma unroll
    for (int q = 0; q < 4; ++q) { v[q] = sh_sc[lane + 32 * q]; m = fmaxf(m, v[q]); }
    for (int off = 16; off > 0; off >>= 1) m = fmaxf(m, __shfl_down(m, off, 32));
    m = __shfl(m, 0, 32);
    float s = 0.0f;
#pragma unroll
    for (int q = 0; q < 4; ++q) { v[q] = expf(v[q] - m); s += v[q]; }
    for (int off = 16; off > 0; off >>= 1) s += __shfl_down(s, off, 32);
    s = __shfl(s, 0, 32);
    float inv = 1.0f / s;
#pragma unroll
    for (int q = 0; q < 4; ++q) sh_sc[lane + 32 * q] = v[q] * inv;
  }
  __syncthreads();

  for (int d = tid; d < 2 * kH; d += 256) {
    const float* te = text_enc + (size_t)b * kT * (2 * kH) + d;
    float acc = 0.0f;
    for (int t = 0; t < kT; ++t) acc += sh_sc[t] * te[(size_t)t * (2 * kH)];
    zout[(size_t)bl * (2 * kH) + d] = acc;
  }
}

// ---------------------------------------------------------------------------
// Output heads. d_out layout: [x (B*L)] [y (B*L)] [o (B*L*2)]
// ---------------------------------------------------------------------------
__global__ void heads_kernel(const float* __restrict__ ys,
                             const float* __restrict__ xyW, const float* __restrict__ xyb,
                             const float* __restrict__ oW, const float* __restrict__ ob,
                             float* __restrict__ out) {
  const int wave = threadIdx.x >> 5, lane = threadIdx.x & 31;
  const int bl = blockIdx.x * 8 + wave;
  float a0 = 0, a1 = 0, a2 = 0, a3 = 0;
  const float* row = ys + (size_t)bl * kH;
  for (int hh = lane; hh < kH; hh += 32) {
    float v = row[hh];
    a0 += v * xyW[hh];       a1 += v * xyW[kH + hh];
    a2 += v * oW[hh];        a3 += v * oW[kH + hh];
  }
  for (int off = 16; off > 0; off >>= 1) {
    a0 += __shfl_down(a0, off, 32); a1 += __shfl_down(a1, off, 32);
    a2 += __shfl_down(a2, off, 32); a3 += __shfl_down(a3, off, 32);
  }
  if (lane == 0) {
    float xv = sigmoidf_(a0 + xyb[0]) * 500.0f;   // X_PAD - 2
    float yv = sigmoidf_(a1 + xyb[1]) * 400.0f;   // Y_PAD - 2
    float o0 = a2 + ob[0], o1 = a3 + ob[1];
    float m = fmaxf(o0, o1);
    float lse = m + logf(expf(o0 - m) + expf(o1 - m));
    const int BL = kB * kL;
    out[bl] = xv;
    out[BL + bl] = yv;
    out[2 * BL + 2 * bl] = o0 - lse;
    out[2 * BL + 2 * bl + 1] = o1 - lse;
  }
}

// ---------------------------------------------------------------------------
// Host orchestration
// ---------------------------------------------------------------------------
extern "C" void kernel_launch(void* const* d_in, const int* in_sizes, int n_in,
                              void* d_out, int out_size, void* d_ws, size_t ws_size,
                              hipStream_t stream) {
  (void)in_sizes; (void)n_in; (void)out_size; (void)ws_size;

  const int*   text_inds = (const int*)d_in[0];
  const int*   clip_inds = (const int*)d_in[1];
  const float* text_emb  = (const float*)d_in[2];
  const float* clip_emb  = (const float*)d_in[3];
  const float* tf_Wih = (const float*)d_in[4],  *tf_Whh = (const float*)d_in[5];
  const float* tf_bih = (const float*)d_in[6],  *tf_bhh = (const float*)d_in[7];
  const float* tb_Wih = (const float*)d_in[8],  *tb_Whh = (const float*)d_in[9];
  const float* tb_bih = (const float*)d_in[10], *tb_bhh = (const float*)d_in[11];
  const float* cf_Wih = (const float*)d_in[12], *cf_Whh = (const float*)d_in[13];
  const float* cf_bih = (const float*)d_in[14], *cf_bhh = (const float*)d_in[15];
  const float* cb_Wih = (const float*)d_in[16], *cb_Whh = (const float*)d_in[17];
  const float* cb_bih = (const float*)d_in[18], *cb_bhh = (const float*)d_in[19];
  const float* Wv = (const float*)d_in[20], *bv = (const float*)d_in[21];
  const float* Wh = (const float*)d_in[22], *bh = (const float*)d_in[23];
  const float* Wu = (const float*)d_in[24];
  const float* dc_Wih = (const float*)d_in[25], *dc_Whh = (const float*)d_in[26];
  const float* dc_bih = (const float*)d_in[27], *dc_bhh = (const float*)d_in[28];
  const float* xyW = (const float*)d_in[29], *xyb = (const float*)d_in[30];
  const float* oW  = (const float*)d_in[31], *ob  = (const float*)d_in[32];
  float* out = (float*)d_out;

  float* ws = (float*)d_ws;
  size_t off = 0;
  auto alloc = [&](size_t n) { float* p = ws + off; off += n; return p; };
  float* Xtext    = alloc((size_t)kB * kT * kH);
  float* Xclip    = alloc((size_t)kB * kL * kH);
  float* xp_tf    = alloc((size_t)kB * kT * 3 * kH);
  float* xp_tb    = alloc((size_t)kB * kT * 3 * kH);
  float* xp_cf    = alloc((size_t)kB * kL * 3 * kH);
  float* xp_cb    = alloc((size_t)kB * kL * 3 * kH);
  float* text_enc = alloc((size_t)kB * kT * 2 * kH);
  float* clip_enc = alloc((size_t)kB * kL * 2 * kH);
  float* o_v      = alloc((size_t)kB * kT * kH);
  float* o_hb     = alloc((size_t)kB * kL * kH);
  float* zbuf     = alloc((size_t)kB * kL * 2 * kH);
  float* dec_in   = alloc((size_t)kB * kL * 4 * kH);
  float* xp_dec   = alloc((size_t)kB * kL * 3 * kH);
  float* dec_ys   = alloc((size_t)kB * kL * kH);
  float* h_ping   = alloc((size_t)kB * kH);
  float* h_pong   = alloc((size_t)kB * kH);

  auto gemm = [&](const float* A, const float* W, const float* bias, float* C,
                  int M, int N, int K) {
    dim3 g(N / 128, M / 64);
    gemm_bias_kernel<<<g, 256, 0, stream>>>(A, W, bias, C, M, N, K);
  };

  // Sequential GRU scan: one fused WMMA+gate kernel per step (4 blocks each).
  auto run_scan = [&](const float* xp, int S, const float* Whh, const float* bhh,
                      float* ys, int ys_stride, int col_off, bool reverse) {
    fill_kernel<<<(kB * kH + 255) / 256, 256, 0, stream>>>(h_ping, kB * kH, 0.0f);
    for (int s = 0; s < S; ++s) {
      int t = reverse ? (S - 1 - s) : s;
      const float* hin = (s & 1) ? h_pong : h_ping;
      float*       hout = (s & 1) ? h_ping : h_pong;
      gru_step_kernel<<<kH / 128, 256, 0, stream>>>(
          xp, hin, hout, Whh, bhh, ys, S, t, ys_stride, col_off);
    }
  };

  // 1) Embedding gathers.
  gather_rows_kernel<<<(kB * kT * kH + 255) / 256, 256, 0, stream>>>(
      text_emb, text_inds, Xtext, kB * kT, kH);
  gather_rows_kernel<<<(kB * kL * kH + 255) / 256, 256, 0, stream>>>(
      clip_emb, clip_inds, Xclip, kB * kL, kH);

  // 2) Batched input projections xp = X @ Wih^T + bih for the 4 encoder GRUs.
  gemm(Xtext, tf_Wih, tf_bih, xp_tf, kB * kT, 3 * kH, kH);
  gemm(Xtext, tb_Wih, tb_bih, xp_tb, kB * kT, 3 * kH, kH);
  gemm(Xclip, cf_Wih, cf_bih, xp_cf, kB * kL, 3 * kH, kH);
  gemm(Xclip, cb_Wih, cb_bih, xp_cb, kB * kL, 3 * kH, kH);

  // 3) Bidirectional GRU scans -> text_enc (B,T,2H), clip_enc (B,L,2H).
  run_scan(xp_tf, kT, tf_Whh, tf_bhh, text_enc, 2 * kH, 0,  false);
  run_scan(xp_tb, kT, tb_Whh, tb_bhh, text_enc, 2 * kH, kH, true);
  run_scan(xp_cf, kL, cf_Whh, cf_bhh, clip_enc, 2 * kH, 0,  false);
  run_scan(xp_cb, kL, cb_Whh, cb_bhh, clip_enc, 2 * kH, kH, true);

  // 4) Attention projections + fused attention.
  gemm(text_enc, Wv, bv, o_v,  kB * kT, kH, 2 * kH);
  gemm(clip_enc, Wh, bh, o_hb, kB * kL, kH, 2 * kH);
  attn_kernel<<<kB * kL, 256, 0, stream>>>(o_hb, o_v, Wu, text_enc, zbuf);

  // 5) Decoder: dec_in = [clip_enc | z], input projection, GRU scan.
  concat_kernel<<<(kB * kL * 4 * kH + 255) / 256, 256, 0, stream>>>(
      clip_enc, zbuf, dec_in, kB * kL, 2 * kH, 2 * kH);
  gemm(dec_in, dc_Wih, dc_bih, xp_dec, kB * kL, 3 * kH, 4 * kH);
  run_scan(xp_dec, kL, dc_Whh, dc_bhh, dec_ys, kH, 0, false);

  // 6) Output heads -> d_out = [x | y | o(log_softmax)].
  heads_kernel<<<(kB * kL) / 8, 256, 0, stream>>>(dec_ys, xyW, xyb, oW, ob, out);
}